// GateGAT_45887430591134
// MI455X (gfx1250) — compile-verified
//
#include <hip/hip_runtime.h>
#include <hip/hip_bf16.h>

typedef __attribute__((ext_vector_type(2))) float v2f;
typedef __attribute__((ext_vector_type(8))) float v8f;

#define LEAKY(x) ((x) > 0.0f ? (x) : 0.01f * (x))

// Ordered-uint encoding of float so unsigned atomicMin/Max implement float min/max.
__device__ __forceinline__ unsigned fenc(float f) {
  unsigned u = __float_as_uint(f);
  return (u & 0x80000000u) ? ~u : (u | 0x80000000u);
}
__device__ __forceinline__ float fdec(unsigned u) {
  unsigned v = (u & 0x80000000u) ? (u & 0x7fffffffu) : ~u;
  return __uint_as_float(v);
}

__global__ void fill_u32(unsigned* __restrict__ p, unsigned v, size_t n) {
  size_t i = (size_t)blockIdx.x * blockDim.x + threadIdx.x;
  size_t stride = (size_t)gridDim.x * blockDim.x;
  for (; i < n; i += stride) p[i] = v;
}

// Pre-pack weights into per-wave B-fragment order for V_WMMA_F32_16X16X4_F32:
//   packB[((ct*(K/4) + s)*32 + lane)*2 + {0,1}] = W[k = 4s + (lane>>4)*2 + {0,1}][col = ct*16 + (lane&15)]
// Generic W indexing: element (k, col) at W[(col>>shift)*head_stride + k*row_stride + (col&mask)]
__global__ void pack_b_kernel(const float* __restrict__ W, float* __restrict__ packB,
                              int ncols, int K, int col_shift, int col_mask,
                              int head_stride, int row_stride) {
  int t = blockIdx.x * blockDim.x + threadIdx.x;
  int total = (ncols >> 4) * (K >> 2) * 32;
  if (t >= total) return;
  int lane = t & 31;
  int s = (t >> 5) % (K >> 2);
  int ct = (t >> 5) / (K >> 2);
  int col = (ct << 4) + (lane & 15);
  int k = (s << 2) + ((lane >> 4) << 1);
  const float* wp = W + (size_t)(col >> col_shift) * (size_t)head_stride + (size_t)(col & col_mask);
  packB[(size_t)t * 2 + 0] = wp[(size_t)k * (size_t)row_stride];
  packB[(size_t)t * 2 + 1] = wp[(size_t)(k + 1) * (size_t)row_stride];
}

// fp32 WMMA GEMM: C[nrows, NCOLS] = act(A[nrows,K]) * W + bias, weights pre-packed.
// One wave32 computes one 16x16 tile; inner loop = 2x global_load_b64 + 1x v_wmma.
// EXACT_ROWS: nrows % 16 == 0 -> no clamp, unpredicated store epilogue.
template <int K, int NCOLS, bool LEAKY_IN, bool EXACT_ROWS>
__global__ void wmma_gemm_f32_kernel(const float* __restrict__ A,
                                     const float* __restrict__ packB,
                                     const float* __restrict__ bias,
                                     float* __restrict__ C, int nrows) {
  const int lane = threadIdx.x & 31;
  const int wave = threadIdx.x >> 5;
  const int wavesPerBlock = blockDim.x >> 5;
  const int rowBase = blockIdx.x << 4;
  const int colTile = blockIdx.y * wavesPerBlock + wave;
  const int colBase = colTile << 4;
  if (colBase >= NCOLS) return;  // uniform per wave -> EXEC all-ones for WMMA

  const int half = lane >> 4;
  const int l16 = lane & 15;
  int arow = rowBase + l16;
  if (!EXACT_ROWS && arow >= nrows) arow = nrows - 1;  // clamp: rows never stored
  // A fragment for step s: v2f at float offset 4s + 2*half  ==  v2f index 2s + half
  const v2f* __restrict__ ap = (const v2f*)(A + (size_t)arow * K) + half;
  // B fragment for step s: packed contiguously, stride 32 v2f per step
  const v2f* __restrict__ bp =
      (const v2f*)packB + (size_t)colTile * (size_t)(K >> 2) * 32 + lane;

  v8f acc = {0.f, 0.f, 0.f, 0.f, 0.f, 0.f, 0.f, 0.f};
#pragma unroll 4
  for (int s = 0; s < (K >> 2); ++s) {
    v2f a = ap[2 * s];
    if (LEAKY_IN) {
      a.x = LEAKY(a.x);
      a.y = LEAKY(a.y);
    }
    v2f b = bp[s * 32];
    acc = __builtin_amdgcn_wmma_f32_16x16x4_f32(false, a, false, b, (short)0, acc,
                                                false, false);
  }
  const int bcol = colBase + l16;
  const float bv = bias ? bias[bcol] : 0.f;
  float* __restrict__ cp = C + (size_t)(rowBase + half * 8) * NCOLS + bcol;
#pragma unroll
  for (int r = 0; r < 8; ++r) {
    if (EXACT_ROWS || (rowBase + r + half * 8) < nrows) cp[(size_t)r * NCOLS] = acc[r] + bv;
  }
}

template <int K, int NCOLS, bool LEAKY_IN>
static void launch_gemm(const float* A, const float* packB, const float* bias, float* C,
                        int nrows, hipStream_t stream) {
  const int rows16 = (nrows + 15) / 16;
  constexpr int colTiles = NCOLS / 16;
  constexpr int wavesPerBlock = colTiles < 4 ? colTiles : 4;
  constexpr int gy = colTiles / wavesPerBlock;
  if ((nrows & 15) == 0) {
    wmma_gemm_f32_kernel<K, NCOLS, LEAKY_IN, true>
        <<<dim3(rows16, gy), dim3(32 * wavesPerBlock), 0, stream>>>(A, packB, bias, C, nrows);
  } else {
    wmma_gemm_f32_kernel<K, NCOLS, LEAKY_IN, false>
        <<<dim3(rows16, gy), dim3(32 * wavesPerBlock), 0, stream>>>(A, packB, bias, C, nrows);
  }
}

// Per-node attention scalars: s_src[n,h] = z[n,h,:].a_src[h,:], same for dst.
__global__ void attn_scores_kernel(const float* __restrict__ z,
                                   const float* __restrict__ a_src,
                                   const float* __restrict__ a_dst,
                                   float* __restrict__ s_src, float* __restrict__ s_dst,
                                   int N, int H, int O) {
  int t = blockIdx.x * blockDim.x + threadIdx.x;
  if (t >= N * H) return;
  int n = t / H, hd = t % H;
  const float* zp = z + (size_t)n * (size_t)(H * O) + (size_t)hd * O;
  const float* as = a_src + (size_t)hd * O;
  const float* ad = a_dst + (size_t)hd * O;
  float ss = 0.f, sd = 0.f;
  for (int o = 0; o < O; ++o) {
    float v = zp[o];
    ss += v * as[o];
    sd += v * ad[o];
  }
  s_src[t] = ss;
  s_dst[t] = sd;
}

// Gate MLP per edge: relu(concat(h1[src],h1[dst]) @ W2 + b2) @ W3 + b3.
__global__ __launch_bounds__(256) void gate_edge_kernel(
    const float* __restrict__ h1, const int* __restrict__ src, const int* __restrict__ dst,
    const float* __restrict__ W2, const float* __restrict__ b2,
    const float* __restrict__ W3, const float* __restrict__ b3,
    float* __restrict__ s_e, unsigned* __restrict__ mm, int E) {
  __shared__ float smin[256], smax[256];
  int t = blockIdx.x * blockDim.x + threadIdx.x;
  float s = 0.f;
  if (t < E) {
    const float* hs = h1 + (size_t)src[t] * 16;
    const float* hd = h1 + (size_t)dst[t] * 16;
    float acc = b3[0];
#pragma unroll
    for (int j = 0; j < 8; ++j) {
      float u = b2[j];
#pragma unroll
      for (int k = 0; k < 16; ++k) u += hs[k] * W2[k * 8 + j];
#pragma unroll
      for (int k = 0; k < 16; ++k) u += hd[k] * W2[(16 + k) * 8 + j];
      u = fmaxf(u, 0.f);
      acc += u * W3[j];
    }
    s = acc;
    s_e[t] = s;
  }
  smin[threadIdx.x] = (t < E) ? s : __builtin_inff();
  smax[threadIdx.x] = (t < E) ? s : -__builtin_inff();
  __syncthreads();
  for (int st = blockDim.x >> 1; st > 0; st >>= 1) {
    if ((int)threadIdx.x < st) {
      smin[threadIdx.x] = fminf(smin[threadIdx.x], smin[threadIdx.x + st]);
      smax[threadIdx.x] = fmaxf(smax[threadIdx.x], smax[threadIdx.x + st]);
    }
    __syncthreads();
  }
  if (threadIdx.x == 0) {
    atomicMin(&mm[0], fenc(smin[0]));
    atomicMax(&mm[1], fenc(smax[0]));
  }
}

__global__ void gate_norm_kernel(const float* __restrict__ s_e,
                                 const unsigned* __restrict__ mm,
                                 float* __restrict__ gate, int E) {
  int t = blockIdx.x * blockDim.x + threadIdx.x;
  if (t >= E) return;
  float mn = fdec(mm[0]);
  float mx = fdec(mm[1]);
  gate[t] = (s_e[t] - mn) / (mx - mn);
}

// Attention logit per (edge, head) + segment max over dst via unsigned atomicMax.
__global__ void logit_max_kernel(const float* __restrict__ s_src,
                                 const float* __restrict__ s_dst,
                                 const float* __restrict__ gate,
                                 const int* __restrict__ src, const int* __restrict__ dst,
                                 float* __restrict__ buf, unsigned* __restrict__ menc,
                                 int E, int H) {
  int t = blockIdx.x * blockDim.x + threadIdx.x;
  if (t >= E * H) return;
  int e = t / H, hd = t % H;
  float lg = s_src[(size_t)src[e] * H + hd] + s_dst[(size_t)dst[e] * H + hd];
  lg = LEAKY(lg) * gate[e];
  buf[t] = lg;
  atomicMax(&menc[(size_t)dst[e] * H + hd], fenc(lg));
}

__global__ void exp_sum_kernel(const int* __restrict__ dst,
                               const unsigned* __restrict__ menc,
                               float* __restrict__ buf, float* __restrict__ den,
                               int E, int H) {
  int t = blockIdx.x * blockDim.x + threadIdx.x;
  if (t >= E * H) return;
  int e = t / H, hd = t % H;
  float m = fdec(menc[(size_t)dst[e] * H + hd]);  // finite: this edge contributed
  float ex = expf(buf[t] - m);
  buf[t] = ex;
  atomicAdd(&den[(size_t)dst[e] * H + hd], ex);
}

// out[dst, h, :] += (ex/den) * z[src, h, :], float4 gather + 4 f32 atomic scatters.
__global__ void aggregate_kernel(const float* __restrict__ buf, const float* __restrict__ den,
                                 const float* __restrict__ z, float* __restrict__ out,
                                 const int* __restrict__ src, const int* __restrict__ dst,
                                 long total, int H, int O) {
  long t = (long)blockIdx.x * blockDim.x + threadIdx.x;
  if (t >= total) return;
  const int CH = (H * O) >> 2;  // float4 chunks per edge
  int e = (int)(t / CH);
  int c = (int)(t % CH);
  int off = c << 2;
  int hd = off / O;
  int s = src[e], d = dst[e];
  float alpha = buf[(size_t)e * H + hd] / fmaxf(den[(size_t)d * H + hd], 1e-12f);
  const float4 z4 = *(const float4*)(z + (size_t)s * (size_t)(H * O) + off);
  float* op = out + (size_t)d * (size_t)(H * O) + off;
  atomicAdd(op + 0, alpha * z4.x);
  atomicAdd(op + 1, alpha * z4.y);
  atomicAdd(op + 2, alpha * z4.z);
  atomicAdd(op + 3, alpha * z4.w);
}

// Edge predictor: score[e,c] = concat(hB[src], hB[dst]) @ Wp + bp,  Wp: [128,2]
__global__ void edge_pred_kernel(const float* __restrict__ hB,
                                 const int* __restrict__ src, const int* __restrict__ dst,
                                 const float* __restrict__ Wp, const float* __restrict__ bp,
                                 float* __restrict__ out, int E) {
  int t = blockIdx.x * blockDim.x + threadIdx.x;
  if (t >= E) return;
  const float* hs = hB + (size_t)src[t] * 64;
  const float* hd = hB + (size_t)dst[t] * 64;
  float a0 = bp[0], a1 = bp[1];
#pragma unroll 4
  for (int k = 0; k < 64; ++k) {
    float v = hs[k];
    a0 += v * Wp[2 * k];
    a1 += v * Wp[2 * k + 1];
  }
#pragma unroll 4
  for (int k = 0; k < 64; ++k) {
    float v = hd[k];
    a0 += v * Wp[2 * (64 + k)];
    a1 += v * Wp[2 * (64 + k) + 1];
  }
  out[2 * t] = a0;
  out[2 * t + 1] = a1;
}

extern "C" void kernel_launch(void* const* d_in, const int* in_sizes, int n_in,
                              void* d_out, int out_size, void* d_ws, size_t ws_size,
                              hipStream_t stream) {
  const float* h   = (const float*)d_in[0];
  const int*   src = (const int*)d_in[1];
  const int*   dst = (const int*)d_in[2];
  const float* W1  = (const float*)d_in[3];
  const float* b1  = (const float*)d_in[4];
  const float* W2  = (const float*)d_in[5];
  const float* b2  = (const float*)d_in[6];
  const float* W3  = (const float*)d_in[7];
  const float* b3  = (const float*)d_in[8];
  const float* fc1 = (const float*)d_in[9];
  const float* a1s = (const float*)d_in[10];
  const float* a1d = (const float*)d_in[11];
  const float* fc2 = (const float*)d_in[12];
  const float* a2s = (const float*)d_in[13];
  const float* a2d = (const float*)d_in[14];
  const float* Wp  = (const float*)d_in[15];
  const float* bp  = (const float*)d_in[16];

  const int N = in_sizes[0] / 128;  // 50000
  const int E = in_sizes[1];        // 800000

  // Workspace layout (floats). z2/out2 alias the z1 region (z1 dead by then).
  float* ws = (float*)d_ws;
  size_t o = 0;
  float* z1    = ws + o; o += (size_t)N * 256;
  float* out1  = ws + o; o += (size_t)N * 256;
  float* exbuf = ws + o; o += (size_t)E * 8;
  float* s1s   = ws + o; o += (size_t)N * 8;
  float* s1d   = ws + o; o += (size_t)N * 8;
  unsigned* menc = (unsigned*)(ws + o); o += (size_t)N * 8;
  float* den   = ws + o; o += (size_t)N * 8;
  float* h1    = ws + o; o += (size_t)N * 16;
  float* s_e   = ws + o; o += (size_t)E;
  float* s2s   = ws + o; o += (size_t)N;
  float* s2d   = ws + o; o += (size_t)N;
  unsigned* mm = (unsigned*)(ws + o); o += 2;
  float* pW1   = ws + o; o += (size_t)1 * 32 * 64;    // (16/16)*(128/4)*32*2
  float* pFc1  = ws + o; o += (size_t)16 * 32 * 64;   // (256/16)*(128/4)*32*2
  float* pFc2  = ws + o; o += (size_t)4 * 64 * 64;    // (64/16)*(256/4)*32*2
  float* z2   = z1;                    // [N,64]
  float* out2 = z1 + (size_t)N * 64;   // [N,64] (hB)

  float* score = (float*)d_out;              // [E,2]
  float* gate  = score + (size_t)E * 2;      // [E,1] second tuple output

  auto fillN = [&](unsigned* p, unsigned v, size_t n) {
    size_t b = (n + 255) / 256;
    if (b > 4096) b = 4096;
    fill_u32<<<dim3((unsigned)b), dim3(256), 0, stream>>>(p, v, n);
  };

  // ---- init accumulators ----
  fillN((unsigned*)out1, 0u, (size_t)N * 256);
  fillN(menc, 0u, (size_t)N * 8);            // 0 decodes below -inf -> max-neutral
  fillN((unsigned*)den, 0u, (size_t)N * 8);
  fillN(&mm[0], 0xFFFFFFFFu, 1);             // min-neutral
  fillN(&mm[1], 0x00000000u, 1);             // max-neutral

  // ---- pack weights into WMMA B-fragment order (tiny, one-shot) ----
  // W1 [128,16]: shift=4 mask=15 hs=0 rs=16   | fc1 [8,128,32]: shift=5 mask=31 hs=4096 rs=32
  // fc2 [256,64]: shift=6 mask=63 hs=0 rs=64
  pack_b_kernel<<<dim3((1 * 32 * 32 + 255) / 256), dim3(256), 0, stream>>>(
      W1, pW1, 16, 128, 4, 15, 0, 16);
  pack_b_kernel<<<dim3((16 * 32 * 32 + 255) / 256), dim3(256), 0, stream>>>(
      fc1, pFc1, 256, 128, 5, 31, 4096, 32);
  pack_b_kernel<<<dim3((4 * 64 * 32 + 255) / 256), dim3(256), 0, stream>>>(
      fc2, pFc2, 64, 256, 6, 63, 0, 64);

  // ---- gate MLP node part: h1 = h @ W1 + b1 (WMMA, 16 cols) ----
  launch_gemm<128, 16, false>(h, pW1, b1, h1, N, stream);

  // ---- layer1 GEMM: z1[N,256] = h @ fc1 (WMMA) ----
  launch_gemm<128, 256, false>(h, pFc1, nullptr, z1, N, stream);

  // ---- per-node attention scalars (layer1) ----
  attn_scores_kernel<<<dim3((N * 8 + 255) / 256), dim3(256), 0, stream>>>(
      z1, a1s, a1d, s1s, s1d, N, 8, 32);

  // ---- edge gate + min/max, then normalize ----
  gate_edge_kernel<<<dim3((E + 255) / 256), dim3(256), 0, stream>>>(
      h1, src, dst, W2, b2, W3, b3, s_e, mm, E);
  gate_norm_kernel<<<dim3((E + 255) / 256), dim3(256), 0, stream>>>(s_e, mm, gate, E);

  // ---- layer1 segment softmax + aggregation ----
  logit_max_kernel<<<dim3((E * 8 + 255) / 256), dim3(256), 0, stream>>>(
      s1s, s1d, gate, src, dst, exbuf, menc, E, 8);
  exp_sum_kernel<<<dim3((E * 8 + 255) / 256), dim3(256), 0, stream>>>(
      dst, menc, exbuf, den, E, 8);
  {
    long total = (long)E * 64;  // E * (8*32/4)
    aggregate_kernel<<<dim3((unsigned)((total + 255) / 256)), dim3(256), 0, stream>>>(
        exbuf, den, z1, out1, src, dst, total, 8, 32);
  }

  // ---- layer2 GEMM: z2[N,64] = leaky(out1) @ fc2 (WMMA, leaky fused in A-load) ----
  launch_gemm<256, 64, true>(out1, pFc2, nullptr, z2, N, stream);

  attn_scores_kernel<<<dim3((N + 255) / 256), dim3(256), 0, stream>>>(
      z2, a2s, a2d, s2s, s2d, N, 1, 64);

  // ---- reset accumulators for layer2 (H=1) ----
  fillN(menc, 0u, (size_t)N);
  fillN((unsigned*)den, 0u, (size_t)N);
  fillN((unsigned*)out2, 0u, (size_t)N * 64);

  logit_max_kernel<<<dim3((E + 255) / 256), dim3(256), 0, stream>>>(
      s2s, s2d, gate, src, dst, exbuf, menc, E, 1);
  exp_sum_kernel<<<dim3((E + 255) / 256), dim3(256), 0, stream>>>(
      dst, menc, exbuf, den, E, 1);
  {
    long total = (long)E * 16;  // E * (64/4)
    aggregate_kernel<<<dim3((unsigned)((total + 255) / 256)), dim3(256), 0, stream>>>(
        exbuf, den, z2, out2, src, dst, total, 1, 64);
  }

  // ---- edge predictor ----
  edge_pred_kernel<<<dim3((E + 255) / 256), dim3(256), 0, stream>>>(
      out2, src, dst, Wp, bp, score, E);
}